// SimpleGCN_27530740368060
// MI455X (gfx1250) — compile-verified
//
#include <hip/hip_runtime.h>

#define N_NODES 50000
#define N_EDGES 640000
#define DIM 128
#define TILES_M ((N_NODES + 15) / 16)   // 3125 exact (50000 = 16*3125)

typedef float v2f __attribute__((ext_vector_type(2)));
typedef float v8f __attribute__((ext_vector_type(8)));

// ---- degree with self-loop -------------------------------------------------
__global__ __launch_bounds__(256) void gcn_deg_init(float* __restrict__ deg) {
  int i = blockIdx.x * 256 + threadIdx.x;
  if (i < N_NODES) deg[i] = 1.0f;                 // self-loop counts once
}

__global__ __launch_bounds__(256) void gcn_deg_accum(const long long* __restrict__ ei,
                                                     float* __restrict__ deg) {
  int e = blockIdx.x * 256 + threadIdx.x;
  if (e < N_EDGES) {
    int dst = (int)ei[N_EDGES + e];               // edge_index[1][e]
    atomicAdd(&deg[dst], 1.0f);
  }
}

__global__ __launch_bounds__(256) void gcn_dinv(const float* __restrict__ deg,
                                                float* __restrict__ dinv) {
  int i = blockIdx.x * 256 + threadIdx.x;
  if (i < N_NODES) {
    float d = deg[i];
    dinv[i] = d > 0.f ? rsqrtf(d) : 0.f;
  }
}

// ---- h = x @ W via V_WMMA_F32_16X16X4_F32 ----------------------------------
// Block = 256 threads = 8 waves. W staged in LDS, K-pair interleaved:
//   sW[kp][n*2 + r] = W[2*kp + r][n]   (kp = 0..63, r = 0..1)
// so each B fragment {W[kb][n], W[kb+1][n]} is one aligned ds_load_b64.
// Each wave computes one 16-row M tile across all 128 output columns.
__global__ __launch_bounds__(256) void gcn_gemm(const float* __restrict__ x,
                                                const float* __restrict__ W,
                                                float* __restrict__ h) {
  __shared__ float sW[DIM * DIM];                 // 64 KB, interleaved layout
  int tid = threadIdx.x;
  // 16384 floats / 256 threads = 16 float4 loads per thread
  for (int i = tid; i < DIM * DIM / 4; i += 256) {
    float4 v = ((const float4*)W)[i];
    int k  = i >> 5;                              // row of W (i*4/128)
    int n0 = (i & 31) * 4;                        // starting column
    float* dstp = sW + (k >> 1) * (2 * DIM) + (k & 1);
    dstp[(n0 + 0) * 2] = v.x;
    dstp[(n0 + 1) * 2] = v.y;
    dstp[(n0 + 2) * 2] = v.z;
    dstp[(n0 + 3) * 2] = v.w;
  }
  __syncthreads();

  int wave = tid >> 5;
  int lane = tid & 31;
  int tileM = blockIdx.x * 8 + wave;
  if (tileM >= TILES_M) return;                   // no barriers after this point

  // A fragment layout (16x4 f32): lanes 0-15 -> M=lane, K={k0,k0+1};
  //                               lanes 16-31 -> M=lane-16, K={k0+2,k0+3}.
  int half = lane >> 4;                           // 0 or 1
  int l16  = lane & 15;
  const float* xrow = x + (size_t)(tileM * 16 + l16) * DIM + half * 2;

  v8f acc[8] = {};
  for (int kk = 0; kk < 32; ++kk) {
    int k0 = kk * 4;
    v2f a = *(const v2f*)(xrow + k0);             // 8-byte global load

    // B fragment (4x16 f32): lanes 0-15 -> K={k0,k0+1}, N=l16;
    //                        lanes 16-31 -> K={k0+2,k0+3}, N=l16.
    // K-pair index: kp = kk*2 + half; contiguous v2f in interleaved sW.
    const v2f* wrow = (const v2f*)sW + (kk * 2 + half) * DIM + l16;
    #pragma unroll
    for (int nt = 0; nt < 8; ++nt) {
      v2f b = wrow[nt * 16];                      // single ds_load_b64
      acc[nt] = __builtin_amdgcn_wmma_f32_16x16x4_f32(
          /*neg_a=*/false, a, /*neg_b=*/false, b,
          /*c_mod=*/(short)0, acc[nt], /*reuse_a=*/false, /*reuse_b=*/false);
    }
  }

  // C/D layout: VGPR v -> lanes 0-15: M=v, lanes 16-31: M=8+v; N = l16.
  int rbase = tileM * 16 + half * 8;
  #pragma unroll
  for (int nt = 0; nt < 8; ++nt) {
    #pragma unroll
    for (int v = 0; v < 8; ++v) {
      h[(size_t)(rbase + v) * DIM + nt * 16 + l16] = acc[nt][v];
    }
  }
}

// ---- out = h * dinv^2 (self-loop term; also initializes d_out) -------------
__global__ __launch_bounds__(256) void gcn_self(const float* __restrict__ h,
                                                const float* __restrict__ dinv,
                                                float* __restrict__ out) {
  int i = blockIdx.x * 256 + threadIdx.x;
  if (i < N_NODES * DIM) {
    int n = i >> 7;
    float d = dinv[n];
    out[i] = h[i] * d * d;
  }
}

// ---- edge scatter: one wave32 per edge, float4 per lane --------------------
__global__ __launch_bounds__(256) void gcn_scatter(const long long* __restrict__ ei,
                                                   const float* __restrict__ h,
                                                   const float* __restrict__ dinv,
                                                   float* __restrict__ out) {
  int gid = blockIdx.x * 256 + threadIdx.x;
  int e = gid >> 5;
  int lane = gid & 31;
  if (e >= N_EDGES) return;
  int src = (int)ei[e];
  int dst = (int)ei[N_EDGES + e];
  float norm = dinv[src] * dinv[dst];
  float4 v = ((const float4*)(h + (size_t)src * DIM))[lane];
  float* o = out + (size_t)dst * DIM + lane * 4;
  atomicAdd(o + 0, v.x * norm);
  atomicAdd(o + 1, v.y * norm);
  atomicAdd(o + 2, v.z * norm);
  atomicAdd(o + 3, v.w * norm);
}

// ---- bias + ReLU -----------------------------------------------------------
__global__ __launch_bounds__(256) void gcn_bias_relu(const float* __restrict__ b,
                                                     float* __restrict__ out) {
  int i = blockIdx.x * 256 + threadIdx.x;
  if (i < N_NODES * DIM) {
    float v = out[i] + b[i & (DIM - 1)];
    out[i] = v > 0.f ? v : 0.f;
  }
}

extern "C" void kernel_launch(void* const* d_in, const int* in_sizes, int n_in,
                              void* d_out, int out_size, void* d_ws, size_t ws_size,
                              hipStream_t stream) {
  const float*     x  = (const float*)d_in[0];       // [N, 128] f32
  const long long* ei = (const long long*)d_in[1];   // [2, E] i64
  const float*     W  = (const float*)d_in[2];       // [128, 128] f32
  const float*     b  = (const float*)d_in[3];       // [128] f32
  float* out = (float*)d_out;                        // [N, 128] f32

  // Workspace layout (floats): h [N*128] | dinv [N] | deg [N]
  float* h    = (float*)d_ws;
  float* dinv = h + (size_t)N_NODES * DIM;
  float* deg  = dinv + N_NODES;

  gcn_deg_init <<<(N_NODES + 255) / 256, 256, 0, stream>>>(deg);
  gcn_deg_accum<<<(N_EDGES + 255) / 256, 256, 0, stream>>>(ei, deg);
  gcn_dinv     <<<(N_NODES + 255) / 256, 256, 0, stream>>>(deg, dinv);
  gcn_gemm     <<<(TILES_M + 7) / 8, 256, 0, stream>>>(x, W, h);
  gcn_self     <<<(N_NODES * DIM + 255) / 256, 256, 0, stream>>>(h, dinv, out);
  gcn_scatter  <<<(N_EDGES * 32 + 255) / 256, 256, 0, stream>>>(ei, h, dinv, out);
  gcn_bias_relu<<<(N_NODES * DIM + 255) / 256, 256, 0, stream>>>(b, out);
}